// Zernike_Layer_WithNoOptimization_74517682586564
// MI455X (gfx1250) — compile-verified
//
#include <hip/hip_runtime.h>

typedef __attribute__((ext_vector_type(2))) float v2f;
typedef __attribute__((ext_vector_type(8))) float v8f;

#define NPIX 1048576
#define NPOW 16
#define NBLK 34                 // 136 modes / 4 per WMMA K-step
#define BLOCKS 2048
#define THREADS 256
#define WAVES_PER_BLOCK (THREADS / 32)
#define TOTAL_WAVES (BLOCKS * WAVES_PER_BLOCK)     // 16384
#define NTILES (NPIX / 16)                          // 65536
#define TILES_PER_WAVE (NTILES / TOTAL_WAVES)       // 4

__global__ __launch_bounds__(THREADS)
void zernike_wmma_kernel(const float* __restrict__ rho,
                         const float* __restrict__ rcu_coeff,
                         const float* __restrict__ acu_theta,
                         float* __restrict__ out)
{
    const int lane  = threadIdx.x & 31;
    const int wave  = (int)blockIdx.x * WAVES_PER_BLOCK + (threadIdx.x >> 5);
    const int m     = lane & 15;                // M (power) / N (pixel) index in half-wave
    const int jbase = (lane >> 4) << 1;         // 0 for lanes 0-15, 2 for lanes 16-31

    // ---- A fragments: A[k', s] = rcu_coeff[s, k'] (16x4 f32 tiles), wave-invariant ----
    // Reused by every wave -> keep regular (cached) loads.
    // ISA 32-bit A 16x4 layout: VGPR0 = K0 (lanes 0-15) / K2 (lanes 16-31), VGPR1 = K1/K3.
    v2f afrag[NBLK];
#pragma unroll
    for (int b = 0; b < NBLK; ++b) {
        const int s0 = 4 * b + jbase;
        afrag[b].x = rcu_coeff[(s0 + 0) * NPOW + m];
        afrag[b].y = rcu_coeff[(s0 + 1) * NPOW + m];
    }

    for (int t = 0; t < TILES_PER_WAVE; ++t) {
        const int    tile = wave + t * TOTAL_WAVES;
        const size_t base = (size_t)tile * 16;

        // ---- B fragments: B[s, n] = acu_theta[s, base+n] (4x16 f32 tiles) ----
        // 544 MB streamed exactly once: non-temporal so we don't thrash L2/WGP$.
        // Mirror layout: VGPR0 = row K0 (lanes 0-15) / K2 (lanes 16-31), VGPR1 = K1/K3.
        v2f bfrag[NBLK];
#pragma unroll
        for (int b = 0; b < NBLK; ++b) {
            const size_t s0 = (size_t)(4 * b + jbase);
            const float* p  = acu_theta + s0 * (size_t)NPIX + base + (size_t)m;
            bfrag[b].x = __builtin_nontemporal_load(p);
            bfrag[b].y = __builtin_nontemporal_load(p + NPIX);
        }

        // ---- D[k', n] = sum_s A[k', s] * B[s, n], chained f32 WMMA over K=136 ----
        v8f acc = {0.f, 0.f, 0.f, 0.f, 0.f, 0.f, 0.f, 0.f};
#pragma unroll
        for (int b = 0; b < NBLK; ++b) {
            acc = __builtin_amdgcn_wmma_f32_16x16x4_f32(
                false, afrag[b], false, bfrag[b], (short)0, acc, false, false);
        }

        // ---- Epilogue: fig[n] = sum_{k'=0..15} rho[n]^k' * D[k', n] ----
        // D layout: lane n holds D[v, n] (v=0..7) in lanes 0-15 and D[v+8, n] in lanes 16-31.
        const float r  = __builtin_nontemporal_load(rho + base + (size_t)m);
        const float r2 = r * r;
        const float r4 = r2 * r2;
        const float r8 = r4 * r4;
        float p       = (lane < 16) ? 1.0f : r8;   // lanes 16-31 start at rho^8
        float partial = 0.0f;
#pragma unroll
        for (int v = 0; v < 8; ++v) {
            partial = fmaf(p, acc[v], partial);
            p *= r;
        }
        const float other = __shfl_xor(partial, 16, 32);
        if (lane < 16) {
            __builtin_nontemporal_store(partial + other, out + base + (size_t)m);
        }
    }
}

extern "C" void kernel_launch(void* const* d_in, const int* in_sizes, int n_in,
                              void* d_out, int out_size, void* d_ws, size_t ws_size,
                              hipStream_t stream) {
    const float* rho       = (const float*)d_in[0];   // [NPIX]
    const float* rcu_coeff = (const float*)d_in[1];   // [136, 16]
    const float* acu_theta = (const float*)d_in[2];   // [136, NPIX]
    float*       out       = (float*)d_out;           // [NPIX]
    (void)in_sizes; (void)n_in; (void)out_size; (void)d_ws; (void)ws_size;

    zernike_wmma_kernel<<<dim3(BLOCKS), dim3(THREADS), 0, stream>>>(
        rho, rcu_coeff, acu_theta, out);
}